// ShoeboxToRIR_26474178413317
// MI455X (gfx1250) — compile-verified
//
#include <hip/hip_runtime.h>
#include <hip/hip_bf16.h>
#include <math.h>

// ---------------------------------------------------------------------------
// ShoeboxToRIR for MI455X (gfx1250, wave32)
//   Kernel 1: image-source scatter into LDS RIR (ds_add_f32 atomics).
//     - whole-image cull when ceil(delay) >= RIR_LEN (drops most of 4991 imgs)
//     - sin(pi(frac+n)) = (-1)^n sin(pi frac): 1 v_sin_f32 per image
//     - Hann cos via branch-free Chebyshev recurrence: pure FMA inner loop
//   Kernel 2: 81-tap FIR as banded f32 GEMM, 24x V_WMMA_F32_16X16X4_F32 per
//     16(batch)x16(sample) tile; B fragments pre-packed as float2 in LDS
//     (1 ds_load_b64/step), A as unguarded global_load_b64 on interior tiles.
// ---------------------------------------------------------------------------

#define FS_F       16000.0f
#define C_SND      343.0f
#define RIR_LEN    3968
#define WIN        81
#define PAD_T      40
#define MAXO       15
#define GRID_W     31            // 2*MAXO+1
#define NCUBE      (31 * 31 * 31)
#define PI_F       3.14159265358979323846f
#define DTH        (PI_F / 40.0f)            // hann cos step
#define K2_CHEB    1.99383461043f            // 2*cos(pi/40)
#define KSTEPS     24                        // 96-wide band / K=4

typedef float v2f __attribute__((ext_vector_type(2)));
typedef float v8f __attribute__((ext_vector_type(8)));

// ---------------------------------------------------------------------------
// Kernel 1: one workgroup per batch. Accumulate raw RIR in LDS, write to ws.
// ---------------------------------------------------------------------------
__global__ __launch_bounds__(256) void rir_scatter_kernel(
    const float* __restrict__ in,   // (B,12)
    float* __restrict__ ws,         // (B, RIR_LEN) raw rir
    float* __restrict__ out_toa)    // (B,)
{
    __shared__ float s_rir[RIR_LEN];
    __shared__ float s_plo[3][9];   // t_lo^e, e=0..8
    __shared__ float s_phi[3][9];   // t_hi^e
    __shared__ float s_room[3], s_mic[3], s_src[3];

    const int tid = threadIdx.x;
    const int b   = blockIdx.x;

    if (tid == 0) {
        const float* p = in + b * 12;
        float room[3], mic[3], src[3];
        for (int i = 0; i < 3; ++i) {
            room[i] = p[i];
            mic[i]  = p[3 + i] * room[i];
            src[i]  = p[6 + i] * room[i];
            s_room[i] = room[i]; s_mic[i] = mic[i]; s_src[i] = src[i];
        }
        const float s0 = sqrtf(1.0f - p[9]);
        const float s1 = sqrtf(1.0f - p[10]);
        const float s2 = sqrtf(1.0f - p[11]);
        // a6 = [a0,a0,a0,a0,a1,a2]; t_lo = tr[[0,2,4]], t_hi = tr[[1,3,5]]
        const float tlo[3] = { s0, s0, s1 };
        const float thi[3] = { s0, s0, s2 };
        for (int a = 0; a < 3; ++a) {
            float pl = 1.0f, ph = 1.0f;
            for (int e = 0; e < 9; ++e) {
                s_plo[a][e] = pl; pl *= tlo[a];
                s_phi[a][e] = ph; ph *= thi[a];
            }
        }
        // time-of-arrival (direct path)
        float dx = mic[0] - src[0], dy = mic[1] - src[1], dz = mic[2] - src[2];
        out_toa[b] = (float)PAD_T + FS_F * sqrtf(dx * dx + dy * dy + dz * dz) / C_SND;
    }
    // zero LDS rir (b128 stores)
    for (int i = tid; i < RIR_LEN / 4; i += 256)
        ((float4*)s_rir)[i] = make_float4(0.f, 0.f, 0.f, 0.f);
    __syncthreads();

    const float rm0 = s_room[0], rm1 = s_room[1], rm2 = s_room[2];
    const float mc0 = s_mic[0],  mc1 = s_mic[1],  mc2 = s_mic[2];
    const float sc0 = s_src[0],  sc1 = s_src[1],  sc2 = s_src[2];

    for (int idx = tid; idx < NCUBE; idx += 256) {
        int gx = idx / (GRID_W * GRID_W) - MAXO;
        int rem = idx % (GRID_W * GRID_W);
        int gy = rem / GRID_W - MAXO;
        int gz = rem % GRID_W - MAXO;
        int ax = gx < 0 ? -gx : gx;
        int ay = gy < 0 ? -gy : gy;
        int az = gz < 0 ? -gz : gz;
        if (ax + ay + az > MAXO) continue;   // L1-ball membership

        // attenuation: exp_lo = |floor(g/2)|, exp_hi = |floor((g+1)/2)|
        int elx = (gx >> 1); elx = elx < 0 ? -elx : elx;
        int ely = (gy >> 1); ely = ely < 0 ? -ely : ely;
        int elz = (gz >> 1); elz = elz < 0 ? -elz : elz;
        int ehx = ((gx + 1) >> 1); ehx = ehx < 0 ? -ehx : ehx;
        int ehy = ((gy + 1) >> 1); ehy = ehy < 0 ? -ehy : ehy;
        int ehz = ((gz + 1) >> 1); ehz = ehz < 0 ? -ehz : ehz;
        float att = s_plo[0][elx] * s_phi[0][ehx]
                  * s_plo[1][ely] * s_phi[1][ehy]
                  * s_plo[2][elz] * s_phi[2][ehz];

        // image position: odd -> r*(g+1) - s ; even -> r*g + s  (python mod: g&1)
        float ix = (gx & 1) ? rm0 * (float)(gx + 1) - sc0 : rm0 * (float)gx + sc0;
        float iy = (gy & 1) ? rm1 * (float)(gy + 1) - sc1 : rm1 * (float)gy + sc1;
        float iz = (gz & 1) ? rm2 * (float)(gz + 1) - sc2 : rm2 * (float)gz + sc2;

        float dx = ix - mc0, dy = iy - mc1, dz = iz - mc2;
        float dist = sqrtf(dx * dx + dy * dy + dz * dz);
        dist = fmaxf(dist, 1e-6f);
        float delay = dist * (FS_F / C_SND);
        float fdi   = ceilf(delay);
        int   idel  = (int)fdi;
        if (idel >= RIR_LEN) continue;       // all 81 taps would be dropped

        float amp  = att * __builtin_amdgcn_rcpf(dist);
        float frac = fdi - delay;            // in [0,1)
        int   jmax = min(WIN, RIR_LEN - idel);   // drop-guard hoisted

        // sin(pi*(frac+n)) = (-1)^n * sin(pi*frac)
        float s0 = __sinf(PI_F * frac);
        // hann cos recurrence seeded at j=-1, j=0 -> branch-free body
        float a0 = DTH * (frac - (float)PAD_T);
        float cA = __cosf(a0 - DTH);  // j = -1
        float cB = __cosf(a0);        // j = 0
        float sgn = s0;               // s0 * (-1)^(j-40), j=0 -> +s0

        #pragma unroll 1
        for (int j = 0; j < jmax; ++j) {
            float t  = frac + (float)(j - PAD_T);
            float at = fabsf(t);
            float hann  = (at <= (float)PAD_T) ? fmaf(0.5f, cB, 0.5f) : 0.0f;
            float sincv = (at < 1e-5f) ? 1.0f
                         : sgn * __builtin_amdgcn_rcpf(PI_F * t);
            atomicAdd(&s_rir[idel + j], amp * sincv * hann);  // ds_add_f32
            float cN = K2_CHEB * cB - cA; cA = cB; cB = cN;   // advance cos
            sgn = -sgn;
        }
    }
    __syncthreads();

    // write raw rir (b128 stores)
    float4* wrow = (float4*)(ws + (size_t)b * RIR_LEN);
    const float4* srow = (const float4*)s_rir;
    for (int i = tid; i < RIR_LEN / 4; i += 256) wrow[i] = srow[i];
}

// ---------------------------------------------------------------------------
// Kernel 2: FIR as banded GEMM with V_WMMA_F32_16X16X4_F32.
//   y[b, n0+n] = sum_{kk=0..95} x[b, n0-40+kk] * Bh[kk, n],  Bh[kk,n] = h[kk-n]
//   Block = 256 threads = 8 waves; wave w -> batch tile w; blockIdx.x -> n0.
// ---------------------------------------------------------------------------
__global__ __launch_bounds__(256) void rir_conv_wmma_kernel(
    const float* __restrict__ ws,   // (B, RIR_LEN) raw rir
    float* __restrict__ out)        // (B, RIR_LEN) filtered rir
{
    __shared__ float s_h[WIN];
    __shared__ v2f   s_Bp[KSTEPS * 32];  // per-step per-lane packed B fragment
    __shared__ float s_hsum;

    const int tid = threadIdx.x;

    if (tid < WIN) {
        float n = (float)(tid - PAD_T);
        float x = 0.25f * n;                       // 2*CUTOFF/FS * n
        float px = PI_F * x;
        float sx = (fabsf(x) < 1e-8f) ? 1.0f : __sinf(px) * __builtin_amdgcn_rcpf(px);
        float w  = 0.5f * (1.0f + __cosf(DTH * n));
        s_h[tid] = sx * w;
    }
    __syncthreads();
    if (tid == 0) {
        float acc = 0.0f;
        for (int i = 0; i < WIN; ++i) acc += s_h[i];
        s_hsum = 1.0f / acc;
    }
    __syncthreads();
    if (tid < WIN) s_h[tid] *= s_hsum;
    __syncthreads();
    // pack B fragments: step s, lane l (hi = l>>4, lr = l&15), kk = 4s + 2hi
    for (int i = tid; i < KSTEPS * 32; i += 256) {
        int s  = i >> 5, ln = i & 31;
        int hi = ln >> 4, lr = ln & 15;
        int kk = 4 * s + 2 * hi;
        int d0 = kk - lr, d1 = kk + 1 - lr;
        float b0 = (d0 >= 0 && d0 < WIN) ? s_h[d0] : 0.0f;
        float b1 = (d1 >= 0 && d1 < WIN) ? s_h[d1] : 0.0f;
        s_Bp[i] = (v2f){ b0, b1 };
    }
    __syncthreads();

    const int lane = tid & 31;
    const int wave = tid >> 5;
    const int hi   = lane >> 4;              // 0: lanes 0-15, 1: lanes 16-31
    const int lr   = lane & 15;
    const int batchBase = blockIdx.y * 128 + wave * 16;
    const int n0 = blockIdx.x * 16;

    const float* rowp = ws + (size_t)(batchBase + lr) * RIR_LEN;

    v8f c = {};  // f32 accumulator, 8 VGPRs

    // interior tiles: every A access in [0, RIR_LEN) -> unguarded b64 loads.
    // Uniform branch (n0 is block-uniform): EXEC stays all-ones for WMMA.
    if (n0 >= PAD_T && n0 + 56 <= RIR_LEN) {
        const float* base = rowp + (n0 - PAD_T) + 2 * hi;  // col even -> 8B aligned
        #pragma unroll
        for (int s = 0; s < KSTEPS; ++s) {
            v2f a  = *(const v2f*)(base + 4 * s);     // global_load_b64
            v2f bf = s_Bp[s * 32 + lane];             // ds_load_b64
            c = __builtin_amdgcn_wmma_f32_16x16x4_f32(
                    false, a, false, bf, (short)0, c, false, false);
        }
    } else {
        #pragma unroll
        for (int s = 0; s < KSTEPS; ++s) {
            int col = n0 - PAD_T + 4 * s + 2 * hi;
            float a0 = (col     >= 0 && col     < RIR_LEN) ? rowp[col]     : 0.0f;
            float a1 = (col + 1 >= 0 && col + 1 < RIR_LEN) ? rowp[col + 1] : 0.0f;
            v2f a  = (v2f){ a0, a1 };
            v2f bf = s_Bp[s * 32 + lane];
            c = __builtin_amdgcn_wmma_f32_16x16x4_f32(
                    false, a, false, bf, (short)0, c, false, false);
        }
    }

    // C/D layout: VGPR v, lanes 0-15 -> M=v, N=lane; lanes 16-31 -> M=v+8
    #pragma unroll
    for (int v = 0; v < 8; ++v) {
        int M = v + 8 * hi;
        out[(size_t)(batchBase + M) * RIR_LEN + (n0 + lr)] = c[v];
    }
}

// ---------------------------------------------------------------------------
extern "C" void kernel_launch(void* const* d_in, const int* in_sizes, int n_in,
                              void* d_out, int out_size, void* d_ws, size_t ws_size,
                              hipStream_t stream) {
    const float* in = (const float*)d_in[0];
    const int B = in_sizes[0] / 12;            // 128
    float* out = (float*)d_out;                // rir (B*3968) then toa (B)
    float* ws  = (float*)d_ws;                 // raw rir, B*3968 f32 (~2 MB)

    rir_scatter_kernel<<<dim3(B), dim3(256), 0, stream>>>(
        in, ws, out + (size_t)B * RIR_LEN);

    rir_conv_wmma_kernel<<<dim3(RIR_LEN / 16, B / 128), dim3(256), 0, stream>>>(
        ws, out);
}